// MPNN_3Conv_50818053046296
// MI455X (gfx1250) — compile-verified
//
#include <hip/hip_runtime.h>

typedef float v2f __attribute__((ext_vector_type(2)));
typedef float v8f __attribute__((ext_vector_type(8)));

static constexpr int NNODE = 50000;
static constexpr long NEDGE = 1600000;

__device__ __forceinline__ v8f v8zero() {
  v8f z = {0.f,0.f,0.f,0.f,0.f,0.f,0.f,0.f};
  return z;
}

__device__ __forceinline__ float lrelu(float x) { return x > 0.f ? x : 0.2f*x; }

// D = A(16x4) * B(4x16) + C, fp32 WMMA.
// Layouts (wave32): A: lane(l16)=M, K = vgpr + 2*half ; B: lane(l16)=N, K = vgpr + 2*half ;
// C/D: lane(l16)=N, M = vgpr + 8*half.
__device__ __forceinline__ v8f wmma4(v2f a, v2f b, v8f c) {
  return __builtin_amdgcn_wmma_f32_16x16x4_f32(false, a, false, b, (short)0, c,
                                               false, false);
}

// acc[t][u] += A(32x32)[t-th 16-row tile] @ W[:, u*16 : u*16+16]
// A in LDS row-major stride lda; W in LDS row-major stride 33.
// B fragments are loaded once per K-step and reused for both M-tiles.
__device__ __forceinline__ void gemm2_k32(const float* A, int lda, const float* W,
                                          v8f acc[2][2], int l16, int half) {
  #pragma unroll
  for (int k0 = 0; k0 < 32; k0 += 4) {
    const int kb = k0 + 2*half;
    v2f b0, b1;
    b0.x = W[kb*33 + l16];       b0.y = W[(kb+1)*33 + l16];
    b1.x = W[kb*33 + 16 + l16];  b1.y = W[(kb+1)*33 + 16 + l16];
    #pragma unroll
    for (int t = 0; t < 2; ++t) {
      const float* ap = A + (t*16 + l16)*lda + k0 + 2*half;
      v2f a; a.x = ap[0]; a.y = ap[1];
      acc[t][0] = wmma4(a, b0, acc[t][0]);
      acc[t][1] = wmma4(a, b1, acc[t][1]);
    }
  }
}

// ---------------------------------------------------------------------------
// Edge conv: per wave, one 32-edge tile (2 WMMA M-tiles, B fragments reused).
// h = [x[row] | x[col] | ea0 | ea1 | ea2]  (IN_DIM = 64 + 32*NSRC)
// raw = relu(h@w1 + b1) @ w2 + b2
// nodesum[row] += raw (atomic);  e_out = lrelu(raw)
// ---------------------------------------------------------------------------
template<int NSRC>
__global__ __launch_bounds__(128)
void edge_conv_kernel(const float* __restrict__ x,
                      const float* __restrict__ ea0,
                      const float* __restrict__ ea1,
                      const float* __restrict__ ea2,
                      const int*   __restrict__ rowIdx,
                      const int*   __restrict__ colIdx,
                      const float* __restrict__ w1, const float* __restrict__ b1,
                      const float* __restrict__ w2, const float* __restrict__ b2,
                      float* __restrict__ e_out,
                      float* __restrict__ nodesum)
{
  constexpr int IN_DIM = 64 + 32*NSRC;
  constexpr int LDH = IN_DIM + 2;   // stride ≡ 2 (mod 4) -> conflict-free column reads

  __shared__ float s_w1[IN_DIM*33];
  __shared__ float s_w2[32*33];
  __shared__ float s_b1[32], s_b2[32];
  __shared__ float s_h [4][32*LDH];
  __shared__ float s_h1[4][32*34];
  __shared__ int   s_ri[4][32];
  __shared__ int   s_ci[4][32];

  const int tid = threadIdx.x;
  for (int i = tid; i < IN_DIM*32; i += 128) s_w1[(i>>5)*33 + (i&31)] = w1[i];
  for (int i = tid; i < 32*32;     i += 128) s_w2[(i>>5)*33 + (i&31)] = w2[i];
  if (tid < 32) { s_b1[tid] = b1[tid]; s_b2[tid] = b2[tid]; }

  const int wave = tid >> 5, lane = tid & 31;
  const int l16 = lane & 15, half = lane >> 4;
  const long tileBase = ((long)blockIdx.x*4 + wave) * 32;

  {
    long e = tileBase + lane; if (e >= NEDGE) e = NEDGE - 1;
    s_ri[wave][lane] = rowIdx[e];
    s_ci[wave][lane] = colIdx[e];
  }
  __syncthreads();

  // Gather concatenated edge-input tile into LDS (node rows hit L2: x is 6.4MB).
  float* H = s_h[wave];
  #pragma unroll 1
  for (int r = 0; r < 32; ++r) {
    long e = tileBase + r; if (e >= NEDGE) e = NEDGE - 1;
    const long re = s_ri[wave][r];
    const long ce = s_ci[wave][r];
    for (int c = lane; c < IN_DIM; c += 32) {
      float v;
      if (c < 32)                         v = x[re*32 + c];
      else if (c < 64)                    v = x[ce*32 + (c-32)];
      else if (NSRC >= 1 && c < 96)       v = ea0[e*32 + (c-64)];
      else if (NSRC >= 2 && c < 128)      v = ea1[e*32 + (c-96)];
      else                                v = ea2[e*32 + (c-128)];
      H[r*LDH + c] = v;
    }
  }
  __syncthreads();

  // GEMM1: h @ w1 (+b1, ReLU) -> s_h1
  v8f acc[2][2];
  acc[0][0] = v8zero(); acc[0][1] = v8zero();
  acc[1][0] = v8zero(); acc[1][1] = v8zero();
  #pragma unroll 4
  for (int k0 = 0; k0 < IN_DIM; k0 += 4) {
    const int kb = k0 + 2*half;
    v2f b0, b1v;
    b0.x  = s_w1[kb*33 + l16];       b0.y  = s_w1[(kb+1)*33 + l16];
    b1v.x = s_w1[kb*33 + 16 + l16];  b1v.y = s_w1[(kb+1)*33 + 16 + l16];
    #pragma unroll
    for (int t = 0; t < 2; ++t) {
      const float* ap = H + (t*16 + l16)*LDH + k0 + 2*half;
      v2f a; a.x = ap[0]; a.y = ap[1];
      acc[t][0] = wmma4(a, b0,  acc[t][0]);
      acc[t][1] = wmma4(a, b1v, acc[t][1]);
    }
  }
  float* H1 = s_h1[wave];
  #pragma unroll
  for (int t = 0; t < 2; ++t) {
    #pragma unroll
    for (int u = 0; u < 2; ++u) {
      const int n = u*16 + l16;
      const float bn = s_b1[n];
      #pragma unroll
      for (int j = 0; j < 8; ++j) {
        float v = acc[t][u][j] + bn;
        H1[(t*16 + j + 8*half)*34 + n] = v > 0.f ? v : 0.f;
      }
    }
  }
  __syncthreads();

  // GEMM2: relu(h1) @ w2
  v8f o[2][2];
  o[0][0] = v8zero(); o[0][1] = v8zero();
  o[1][0] = v8zero(); o[1][1] = v8zero();
  gemm2_k32(H1, 34, s_w2, o, l16, half);

  // Scatter pre-activation to node sums; store lrelu() as edge features.
  #pragma unroll
  for (int t = 0; t < 2; ++t) {
    #pragma unroll
    for (int u = 0; u < 2; ++u) {
      const int n = u*16 + l16;
      const float bn = s_b2[n];
      #pragma unroll
      for (int j = 0; j < 8; ++j) {
        const int m = t*16 + j + 8*half;
        const long e = tileBase + m;
        if (e < NEDGE) {
          const float raw = o[t][u][j] + bn;
          unsafeAtomicAdd(&nodesum[(long)s_ri[wave][m]*32 + n], raw);
          e_out[e*32 + n] = lrelu(raw);
        }
      }
    }
  }
}

// ---------------------------------------------------------------------------
// 3-layer 32->32 MLP (lrelu after layers 1,2), per-wave 32-row tiles.
// Supports in == out (tile fully staged to LDS before writeback).
// ---------------------------------------------------------------------------
__global__ __launch_bounds__(128)
void mlp3_kernel(const float* in,
                 const float* __restrict__ w1, const float* __restrict__ b1,
                 const float* __restrict__ w2, const float* __restrict__ b2,
                 const float* __restrict__ w3, const float* __restrict__ b3,
                 float* out, long R)
{
  __shared__ float s_w[3][32*33];
  __shared__ float s_b[3][32];
  __shared__ float s_t[4][32*34];
  __shared__ float s_u[4][32*34];

  const int tid = threadIdx.x;
  for (int i = tid; i < 1024; i += 128) {
    const int r = i >> 5, c = i & 31;
    s_w[0][r*33 + c] = w1[i];
    s_w[1][r*33 + c] = w2[i];
    s_w[2][r*33 + c] = w3[i];
  }
  if (tid < 32) { s_b[0][tid] = b1[tid]; s_b[1][tid] = b2[tid]; s_b[2][tid] = b3[tid]; }

  const int wave = tid >> 5, lane = tid & 31;
  const int l16 = lane & 15, half = lane >> 4;
  const long rowBase = ((long)blockIdx.x*4 + wave) * 32;

  float* T = s_t[wave];
  float* U = s_u[wave];
  for (int i = lane; i < 1024; i += 32) {
    const int r = i >> 5, c = i & 31;
    const long gr = rowBase + r;
    T[r*34 + c] = (gr < R) ? in[gr*32 + c] : 0.f;
  }
  __syncthreads();

  // layer 1 -> U (lrelu)
  v8f a0[2][2];
  a0[0][0] = v8zero(); a0[0][1] = v8zero(); a0[1][0] = v8zero(); a0[1][1] = v8zero();
  gemm2_k32(T, 34, s_w[0], a0, l16, half);
  #pragma unroll
  for (int t = 0; t < 2; ++t)
    #pragma unroll
    for (int u = 0; u < 2; ++u) {
      const int n = u*16 + l16; const float bn = s_b[0][n];
      #pragma unroll
      for (int j = 0; j < 8; ++j)
        U[(t*16 + j + 8*half)*34 + n] = lrelu(a0[t][u][j] + bn);
    }
  __syncthreads();

  // layer 2 -> T (lrelu)
  v8f a1[2][2];
  a1[0][0] = v8zero(); a1[0][1] = v8zero(); a1[1][0] = v8zero(); a1[1][1] = v8zero();
  gemm2_k32(U, 34, s_w[1], a1, l16, half);
  #pragma unroll
  for (int t = 0; t < 2; ++t)
    #pragma unroll
    for (int u = 0; u < 2; ++u) {
      const int n = u*16 + l16; const float bn = s_b[1][n];
      #pragma unroll
      for (int j = 0; j < 8; ++j)
        T[(t*16 + j + 8*half)*34 + n] = lrelu(a1[t][u][j] + bn);
    }
  __syncthreads();

  // layer 3 -> out (no activation)
  v8f a2[2][2];
  a2[0][0] = v8zero(); a2[0][1] = v8zero(); a2[1][0] = v8zero(); a2[1][1] = v8zero();
  gemm2_k32(T, 34, s_w[2], a2, l16, half);
  #pragma unroll
  for (int t = 0; t < 2; ++t)
    #pragma unroll
    for (int u = 0; u < 2; ++u) {
      const int n = u*16 + l16; const float bn = s_b[2][n];
      #pragma unroll
      for (int j = 0; j < 8; ++j) {
        const long gr = rowBase + t*16 + j + 8*half;
        if (gr < R) out[gr*32 + n] = a2[t][u][j] + bn;
      }
    }
}

// ---------------------------------------------------------------------------
// Small utility kernels
// ---------------------------------------------------------------------------
__global__ void zero_kernel(float* __restrict__ p, long n) {
  const long i = (long)blockIdx.x * blockDim.x + threadIdx.x;
  if (i < n) p[i] = 0.f;
}

__global__ void degree_kernel(const int* __restrict__ rowIdx, float* __restrict__ cnt) {
  const long i = (long)blockIdx.x * blockDim.x + threadIdx.x;
  if (i < NEDGE) unsafeAtomicAdd(&cnt[rowIdx[i]], 1.0f);
}

__global__ void node_finalize(const float* __restrict__ nodesum,
                              const float* __restrict__ cnt,
                              float* __restrict__ xout) {
  const long i = (long)blockIdx.x * blockDim.x + threadIdx.x;
  if (i < (long)NNODE*32) {
    float c = cnt[i >> 5];
    c = c < 1.f ? 1.f : c;
    xout[i] = lrelu(nodesum[i] / c);
  }
}

// ---------------------------------------------------------------------------
extern "C" void kernel_launch(void* const* d_in, const int* in_sizes, int n_in,
                              void* d_out, int out_size, void* d_ws, size_t ws_size,
                              hipStream_t stream)
{
  const float* node_attr = (const float*)d_in[0];
  const float* edge_attr = (const float*)d_in[1];
  const int*   edge_index = (const int*)d_in[2];
  const float* c1w1 = (const float*)d_in[3];  const float* c1b1 = (const float*)d_in[4];
  const float* c1w2 = (const float*)d_in[5];  const float* c1b2 = (const float*)d_in[6];
  const float* c2w1 = (const float*)d_in[7];  const float* c2b1 = (const float*)d_in[8];
  const float* c2w2 = (const float*)d_in[9];  const float* c2b2 = (const float*)d_in[10];
  const float* c3w1 = (const float*)d_in[11]; const float* c3b1 = (const float*)d_in[12];
  const float* c3w2 = (const float*)d_in[13]; const float* c3b2 = (const float*)d_in[14];
  const float* nw1  = (const float*)d_in[15]; const float* nb1  = (const float*)d_in[16];
  const float* nw2  = (const float*)d_in[17]; const float* nb2  = (const float*)d_in[18];
  const float* nw3  = (const float*)d_in[19]; const float* nb3  = (const float*)d_in[20];
  const float* ew1  = (const float*)d_in[21]; const float* eb1  = (const float*)d_in[22];
  const float* ew2  = (const float*)d_in[23]; const float* eb2  = (const float*)d_in[24];
  const float* ew3  = (const float*)d_in[25]; const float* eb3  = (const float*)d_in[26];

  const int* row = edge_index;
  const int* col = edge_index + NEDGE;

  // workspace layout
  float* e1      = (float*)d_ws;                 // E*32
  float* e2      = e1 + NEDGE*32;                // E*32
  float* nodesum = e2 + NEDGE*32;                // N*32
  float* cnt     = nodesum + (long)NNODE*32;     // N
  float* xcur    = cnt + NNODE;                  // N*32

  float* out_node = (float*)d_out;               // N*32
  float* out_edge = out_node + (long)NNODE*32;   // E*32

  const int convGrid = (int)((NEDGE + 127) / 128);         // 12500
  const long nsumLen = (long)NNODE*32;
  const int nzGrid   = (int)((nsumLen + NNODE + 255) / 256);
  const int nsGrid   = (int)((nsumLen + 255) / 256);
  const int degGrid  = (int)((NEDGE + 255) / 256);

  // degree (row counts), shared by all three convs; zero nodesum+cnt together
  zero_kernel<<<nzGrid, 256, 0, stream>>>(nodesum, nsumLen + NNODE);
  degree_kernel<<<degGrid, 256, 0, stream>>>(row, cnt);

  // ---- conv 1: in = [x|x|edge_attr] (96) ----
  edge_conv_kernel<1><<<convGrid, 128, 0, stream>>>(
      node_attr, edge_attr, nullptr, nullptr, row, col,
      c1w1, c1b1, c1w2, c1b2, e1, nodesum);
  node_finalize<<<nsGrid, 256, 0, stream>>>(nodesum, cnt, xcur);
  zero_kernel<<<nsGrid, 256, 0, stream>>>(nodesum, nsumLen);

  // ---- conv 2: in = [x|x|e1|edge_attr] (128) ----
  edge_conv_kernel<2><<<convGrid, 128, 0, stream>>>(
      xcur, e1, edge_attr, nullptr, row, col,
      c2w1, c2b1, c2w2, c2b2, e2, nodesum);
  node_finalize<<<nsGrid, 256, 0, stream>>>(nodesum, cnt, xcur);
  zero_kernel<<<nsGrid, 256, 0, stream>>>(nodesum, nsumLen);

  // ---- conv 3: in = [x|x|e2|e1|edge_attr] (160); e3 -> d_out edge region ----
  edge_conv_kernel<3><<<convGrid, 128, 0, stream>>>(
      xcur, e2, e1, edge_attr, row, col,
      c3w1, c3b1, c3w2, c3b2, out_edge, nodesum);
  node_finalize<<<nsGrid, 256, 0, stream>>>(nodesum, cnt, xcur);

  // ---- output MLPs (WMMA), edge MLP runs in-place on out_edge ----
  mlp3_kernel<<<(int)((NNODE + 127) / 128), 128, 0, stream>>>(
      xcur, nw1, nb1, nw2, nb2, nw3, nb3, out_node, (long)NNODE);
  mlp3_kernel<<<convGrid, 128, 0, stream>>>(
      out_edge, ew1, eb1, ew2, eb2, ew3, eb3, out_edge, NEDGE);
}